// MyGNNLayer_1443109011542
// MI455X (gfx1250) — compile-verified
//
#include <hip/hip_runtime.h>

typedef __attribute__((ext_vector_type(2))) float v2f;
typedef __attribute__((ext_vector_type(4))) float v4f;
typedef __attribute__((ext_vector_type(8))) float v8f;

// D = A(16x4 f32) x B(4x16 f32) + C(16x16 f32), wave32
#define WMMA_F32_K4(Aop, Bop, Cop) \
  __builtin_amdgcn_wmma_f32_16x16x4_f32(false, (Aop), false, (Bop), (short)0, (Cop), false, false)

constexpr int Bn = 64, Nn = 512, Fn = 64, On = 128;

// ---------------------------------------------------------------------------
// Kernel A: per-batch pairwise gaussian stats via f32 WMMA Gram tiles.
//   G[i,j] = c_i . c_j (K=4)   dist = |c_i|^2 + |c_j|^2 - 2G
//   t1[m] = mask*w_sum - 1 ; t2[m] = mask*dw_sum
// grid = (64 batches, 4 j-tile groups), block = 256 (8 waves, 1 j-tile/wave)
// ---------------------------------------------------------------------------
__global__ __launch_bounds__(256) void pairwise_stats_kernel(
    const float* __restrict__ x, const float* __restrict__ mask,
    float* __restrict__ t1, float* __restrict__ t2)
{
  __shared__ __align__(16) float lds_c[Nn * 4];   // coords, row-major [j][k]
  __shared__ __align__(16) float lds_n[Nn];       // squared norms

  const int b   = blockIdx.x;
  const int tid = threadIdx.x;

  // Stage last-4 coords of each of the 512 rows + norms (2 rows per thread)
  for (int j = tid; j < Nn; j += 256) {
    v4f c = *(const v4f*)&x[((size_t)b * Nn + j) * Fn + (Fn - 4)];
    *(v4f*)&lds_c[j * 4] = c;
    lds_n[j] = c.x * c.x + c.y * c.y + c.z * c.z + c.w * c.w;
  }
  __syncthreads();

  const int lane = tid & 31;
  const int wave = tid >> 5;
  const int jt   = blockIdx.y * 8 + wave;  // 0..31: this wave's 16-column tile
  const int nn   = lane & 15;
  const int hi   = lane >> 4;

  // B operand (4x16): B[k,n] = c[jt*16+n][k]; lane = n + 16*(k>>1), vgpr = k&1
  v2f bop = *(const v2f*)&lds_c[(jt * 16 + nn) * 4 + 2 * hi];
  const float nj = lds_n[jt * 16 + nn];

  float swsum = 0.f, sdsum = 0.f;
  for (int it = 0; it < 32; ++it) {
    // A operand (16x4): A[m,k] = c[it*16+m][k]; same lane pattern as B
    v2f aop = *(const v2f*)&lds_c[(it * 16 + nn) * 4 + 2 * hi];
    v8f g = {};
    g = WMMA_F32_K4(aop, bop, g);  // v_wmma_f32_16x16x4_f32

    // row norms for this lane's 8 rows: i = it*16 + 8*hi + r
    v4f ni0 = *(const v4f*)&lds_n[it * 16 + 8 * hi];
    v4f ni1 = *(const v4f*)&lds_n[it * 16 + 8 * hi + 4];
#pragma unroll
    for (int r = 0; r < 8; ++r) {
      float ni   = (r < 4) ? ni0[r] : ni1[r - 4];
      float dist = ni + nj - 2.f * g[r];
      float w    = __expf(-10.f * dist);
      swsum += w;
      sdsum += dist * w;
    }
  }
  // lanes (n,hi=0) and (n,hi=1) each summed half the i rows -> combine
  swsum += __shfl_xor(swsum, 16, 32);
  sdsum += __shfl_xor(sdsum, 16, 32);

  if (hi == 0) {  // after last WMMA; safe to diverge
    const int    j  = jt * 16 + nn;
    const size_t m  = (size_t)b * Nn + j;
    const float  mk = mask[m];
    t1[m] = mk * swsum - 1.f;
    t2[m] = mk * sdsum;
  }
}

// ---------------------------------------------------------------------------
// Kernel B: out[m,:] = mask[m]*( X@A0 + t1[m]*(X@A1) + t2[m]*A[128,:] + b )
// grid = 2048 M-tiles of 16 rows; block = 256 (8 waves = 8 column tiles of 16)
// Two f32 WMMA accumulator chains, 16 K=4 steps each (K=64).
// ---------------------------------------------------------------------------
__global__ __launch_bounds__(256) void gemm_fused_kernel(
    const float* __restrict__ x, const float* __restrict__ Aw,
    const float* __restrict__ bias, const float* __restrict__ mask,
    const float* __restrict__ t1, const float* __restrict__ t2,
    float* __restrict__ out)
{
  __shared__ __align__(16) float lds_x[16 * Fn];  // 16-row X tile (4KB)
  __shared__ float lds_m[16], lds_t1[16], lds_t2[16];

  const int m0  = blockIdx.x * 16;
  const int tid = threadIdx.x;

  // cooperative stage: 1024 floats = 256 threads x float4
  *(v4f*)&lds_x[tid * 4] = *(const v4f*)&x[(size_t)m0 * Fn + tid * 4];
  if (tid < 16) {
    lds_m[tid]  = mask[m0 + tid];
    lds_t1[tid] = t1[m0 + tid];
    lds_t2[tid] = t2[m0 + tid];
  }
  __syncthreads();

  const int lane = tid & 31;
  const int wave = tid >> 5;
  const int c0   = wave * 16;
  const int nn   = lane & 15;
  const int hi   = lane >> 4;
  const int col  = c0 + nn;

  v8f acc0 = {}, acc1 = {};
#pragma unroll
  for (int kc = 0; kc < 16; ++kc) {
    const int k = kc * 4 + 2 * hi;
    // A operand: rows of X tile; A[m,k] at lane = m + 16*(k>>1)
    v2f av = *(const v2f*)&lds_x[nn * Fn + k];
    // B operands: weight columns, chain0 = A[0:64], chain1 = A[64:128]
    v2f b0 = { Aw[(size_t)k * On + col],        Aw[(size_t)(k + 1) * On + col] };
    v2f b1 = { Aw[(size_t)(k + 64) * On + col], Aw[(size_t)(k + 65) * On + col] };
    acc0 = WMMA_F32_K4(av, b0, acc0);
    acc1 = WMMA_F32_K4(av, b1, acc1);
  }

  const float a128 = Aw[(size_t)128 * On + col];
  const float bb   = bias[col];
#pragma unroll
  for (int r = 0; r < 8; ++r) {
    const int   mr = r + 8 * hi;                 // row within tile (C/D layout)
    const float o  = lds_m[mr] *
        (acc0[r] + lds_t1[mr] * acc1[r] + lds_t2[mr] * a128 + bb);
    out[(size_t)(m0 + mr) * On + col] = o;
  }
}

extern "C" void kernel_launch(void* const* d_in, const int* in_sizes, int n_in,
                              void* d_out, int out_size, void* d_ws, size_t ws_size,
                              hipStream_t stream)
{
  const float* x    = (const float*)d_in[0];  // (64,512,64)
  const float* mask = (const float*)d_in[1];  // (64,512)
  const float* Aw   = (const float*)d_in[2];  // (129,128)
  const float* bias = (const float*)d_in[3];  // (128,)
  float*       out  = (float*)d_out;          // (64,512,128)

  float* t1 = (float*)d_ws;                   // 32768 floats
  float* t2 = t1 + (size_t)Bn * Nn;           // 32768 floats (256KB total)

  dim3 gridA(Bn, 4);
  pairwise_stats_kernel<<<gridA, 256, 0, stream>>>(x, mask, t1, t2);

  gemm_fused_kernel<<<(Bn * Nn) / 16, 256, 0, stream>>>(
      x, Aw, bias, mask, t1, t2, out);
}